// DifferentiableMPCLayer_28698971472013
// MI455X (gfx1250) — compile-verified
//
#include <hip/hip_runtime.h>
#include <hip/hip_bf16.h>
#include <stdint.h>

#define H      10
#define NR     36
#define NPAIR  (NR / 2)
#define ITERS  15
#define BLK    128

typedef float v2f __attribute__((ext_vector_type(2)));

__device__ __forceinline__ float ftanh(float x) {
#if __has_builtin(__builtin_amdgcn_tanhf)
  // gfx1250-only hardware transcendental: v_tanh_f32
  return __builtin_amdgcn_tanhf(x);
#else
  float e = __expf(2.0f * x);
  return 1.0f - 2.0f / (e + 1.0f);
#endif
}

// atan2 via v_rcp_f32 + odd minimax poly (~1e-6), avoids IEEE divide sequence.
__device__ __forceinline__ float fatan2(float y, float x) {
  float ax = fabsf(x), ay = fabsf(y);
  float mx = fmaxf(ax, ay), mn = fminf(ax, ay);
  float a  = mn * __builtin_amdgcn_rcpf(mx);
  float s  = a * a;
  float p  = -0.0117212f;
  p = fmaf(p, s,  0.05265332f);
  p = fmaf(p, s, -0.11643287f);
  p = fmaf(p, s,  0.19354346f);
  p = fmaf(p, s, -0.33262347f);
  p = fmaf(p, s,  0.99997726f);
  float r = p * a;                       // atan(mn/mx) in [0, pi/4]
  r = (ay > ax) ? (1.57079632679f - r) : r;
  r = (x < 0.0f) ? (3.14159265359f - r) : r;
  return (y < 0.0f) ? -r : r;
}

__global__ __launch_bounds__(BLK) void mpc_kernel(
    const float* __restrict__ init_state,  // (B,3)
    const float* __restrict__ scan,        // (B,36)
    const float* __restrict__ target,      // (B,2)
    const float* __restrict__ wdist,       // (B,1)
    const float* __restrict__ wobs,        // (B,1)
    const float* __restrict__ wangle,      // (B,1)
    const float* __restrict__ wrv,         // (B,1)
    const float* __restrict__ wrw,         // (B,1)
    float* __restrict__ out,               // (B,2)
    int B, float upd_scale)                // upd_scale = LR_INNER / B (grad of mean)
{
  __shared__ float  s_scan[BLK * NR];      // staged raw scan rows, [row][ray]
  __shared__ float4 s_obs[NPAIR * BLK];    // NEGATED (-ox0,-ox1,-oy0,-oy1) per ray-pair, [pair][lane]

  const int tid = threadIdx.x;
  const int i0  = blockIdx.x * BLK;
  const int i   = i0 + tid;

  // ---- Stage this block's scan rows into LDS.
  // Full blocks: CDNA5 async DMA global->LDS (ASYNCcnt path), 9 x b128 per lane,
  // fully coalesced (the 128x36 f32 region is contiguous).
  if (i0 + BLK <= B) {
    const char* gb   = (const char*)scan + (size_t)i0 * NR * 4;
    unsigned    lds0 = (unsigned)(uintptr_t)(void*)&s_scan[0];
#pragma unroll
    for (int k = 0; k < (BLK * NR) / (BLK * 4); ++k) {   // 9 chunks of 16B/lane
      unsigned off = (unsigned)((k * BLK + tid) * 16);
      asm volatile("global_load_async_to_lds_b128 %0, %1, %2"
                   :: "v"(lds0 + off), "v"(off), "s"(gb) : "memory");
    }
    asm volatile("s_wait_asynccnt 0" ::: "memory");
  } else if (i < B) {                      // ragged tail (not hit for B=16384)
    for (int j = 0; j < NR; ++j)
      s_scan[tid * NR + j] = scan[(size_t)i * NR + j];
  }
  __syncthreads();

  if (i >= B) return;

  // ---- Build per-lane NEGATED obstacle pair tile: obs_j = scan_j * (cos th_j, sin th_j),
  // th_j = j * 2*pi/35 (linspace(0, 2pi, 36), endpoint included).
  // Stored negated so the packed distance kernel needs no VOP3P neg modifiers.
  const float STEP = 0.17951958020513104f;
#pragma unroll 3
  for (int k = 0; k < NPAIR; ++k) {
    float sc0 = s_scan[tid * NR + 2 * k + 0];
    float sc1 = s_scan[tid * NR + 2 * k + 1];
    float s0, c0, s1, c1;
    __sincosf((float)(2 * k)     * STEP, &s0, &c0);
    __sincosf((float)(2 * k + 1) * STEP, &s1, &c1);
    s_obs[k * BLK + tid] = make_float4(-sc0 * c0, -sc1 * c1, -sc0 * s0, -sc1 * s1);
  }
  // s_obs column [.,tid] is written and read only by this thread: no barrier needed.

  // ---- Per-robot inputs (weights pre-folded with cost constants).
  const float x0   = init_state[(size_t)i * 3 + 0];
  const float y0   = init_state[(size_t)i * 3 + 1];
  const float yaw0 = init_state[(size_t)i * 3 + 2];
  const float tx   = target[(size_t)i * 2 + 0];
  const float ty   = target[(size_t)i * 2 + 1];
  const float wd20  = 20.0f  * wdist[i];   // d(10*wd*dist^2)/dpos factor
  const float wo400 = 400.0f * wobs[i];    // d(100*wo*u^4)/du factor
  const float wa10  = 10.0f  * wangle[i];  // d(5*wa*aerr^2)/daerr factor
  const float wv2   = 2.0f   * wrv[i];     // d(wv*v^2)/dv
  const float ww02  = 0.2f   * wrw[i];     // d(0.1*ww*w^2)/dw

  const float DT = 0.1f, MAXV = 0.22f, MAXW = 2.8f;
  const float TWO_PI = 6.283185307179586f, INV_TWO_PI = 0.15915494309189535f;

  float a0[H], a1[H];
#pragma unroll
  for (int t = 0; t < H; ++t) { a0[t] = 0.0f; a1[t] = 0.0f; }

  float xs[H], ys[H], yw[H];               // saved post-step states (VGPR-resident)

  for (int it = 0; it < ITERS; ++it) {
    // ---------- forward rollout (cost value itself is never needed) ----------
    float x = x0, y = y0, yaw = yaw0;
#pragma unroll
    for (int t = 0; t < H; ++t) {
      float v = (ftanh(a0[t]) + 1.0f) * (0.5f * MAXV);
      float w = ftanh(a1[t]) * MAXW;
      float sy_, cy_;
      __sincosf(yaw, &sy_, &cy_);
      x   += v * cy_ * DT;
      y   += v * sy_ * DT;
      yaw += w * DT;
      xs[t] = x; ys[t] = y; yw[t] = yaw;
    }
    // ---------- analytic reverse sweep ----------
    float gx = 0.0f, gy = 0.0f, gyaw = 0.0f;
#pragma unroll
    for (int t = H - 1; t >= 0; --t) {
      float px = xs[t], py = ys[t], yp = yw[t];
      float psi = (t == 0) ? yaw0 : yw[t - 1];
      float dx = tx - px, dy = ty - py;
      // goal distance term: += wd20*(px-tx) == -= wd20*dx (bit-exact)
      gx -= wd20 * dx;
      gy -= wd20 * dy;
      // heading-error term; atan2(sin,cos) wrap has unit derivative
      float r2 = dx * dx + dy * dy;
      float th = fatan2(dy, dx) - yp;
      float aerr = th - TWO_PI * rintf(th * INV_TWO_PI);
      float ga = wa10 * aerr;
      gyaw -= ga;
      float ir2 = __builtin_amdgcn_rcpf(r2);
      gx += ga * dy * ir2;
      gy -= ga * dx * ir2;
      // collision term: argmin over obstacle pairs.  Packed-f32 VOP3P distance
      // math (v_pk_add/mul/fma_f32) + index-in-mantissa uint min-reduction:
      // d2 >= 0 so its bit pattern orders as u32; low 6 mantissa bits carry the
      // ray index (v_and_or_b32 + v_min_u32).  Exact d2 recomputed after decode.
      v2f px2 = {px, px}, py2 = {py, py};
      unsigned acca = 0x7F7FFFFFu, accb = 0x7F7FFFFFu;   // FLT_MAX pattern
#pragma unroll 3
      for (int k = 0; k < NPAIR; ++k) {
        float4 o = s_obs[k * BLK + tid];       // ds_load_b128: 2 negated obstacles
        v2f nox = {o.x, o.y};
        v2f noy = {o.z, o.w};
        v2f ddx, ddy, dd2, d2;
        asm("v_pk_add_f32 %0, %1, %2" : "=v"(ddx) : "v"(px2), "v"(nox));    // px - ox
        asm("v_pk_add_f32 %0, %1, %2" : "=v"(ddy) : "v"(py2), "v"(noy));    // py - oy
        asm("v_pk_mul_f32 %0, %1, %1" : "=v"(dd2) : "v"(ddx));              // ddx^2
        asm("v_pk_fma_f32 %0, %1, %1, %2" : "=v"(d2) : "v"(ddy), "v"(dd2)); // + ddy^2
        unsigned e0 = (__float_as_uint(d2.x) & 0xFFFFFFC0u) | (unsigned)(2 * k);
        unsigned e1 = (__float_as_uint(d2.y) & 0xFFFFFFC0u) | (unsigned)(2 * k + 1);
        acca = (e0 < acca) ? e0 : acca;        // v_min_u32
        accb = (e1 < accb) ? e1 : accb;        // v_min_u32
      }
      unsigned acc = (acca < accb) ? acca : accb;
      int jmin = (int)(acc & 63u);
      float4 ob = s_obs[(jmin >> 1) * BLK + tid];   // conflict-free: stride 2048B = 0 mod 64 banks
      float nx = (jmin & 1) ? ob.y : ob.x;          // negated ox*, oy*
      float ny = (jmin & 1) ? ob.w : ob.z;
      float bdx = px + nx;                          // exact px - ox*
      float bdy = py + ny;
      float m2  = fmaf(bdy, bdy, bdx * bdx);        // exact min squared distance
      float md = __builtin_amdgcn_sqrtf(m2);
      float u  = fmaxf(0.4f - md, 0.0f);
      float coef = -wo400 * u * u * u * __builtin_amdgcn_rcpf(md);
      gx += coef * bdx;
      gy += coef * bdy;
      // control-effort terms + dynamics adjoint
      float th0 = ftanh(a0[t]);
      float th1 = ftanh(a1[t]);
      float v = (th0 + 1.0f) * (0.5f * MAXV);
      float w = th1 * MAXW;
      float gv = wv2 * v;
      float gw = ww02 * w + gyaw * DT;
      float sp, cp;
      __sincosf(psi, &sp, &cp);
      gv += (gx * cp + gy * sp) * DT;
      float gpsi = gyaw + (gy * cp - gx * sp) * (v * DT);
      // chain through tanh squashing; upd_scale carries LR/B (cost.mean())
      a0[t] -= upd_scale * gv * (0.5f * MAXV) * (1.0f - th0 * th0);
      a1[t] -= upd_scale * gw * MAXW * (1.0f - th1 * th1);
      gyaw = gpsi;   // gx, gy carry unchanged (dx'/dx = 1)
    }
  }

  float v0 = (ftanh(a0[0]) + 1.0f) * (0.5f * MAXV);
  float w0 = ftanh(a1[0]) * MAXW;
  out[(size_t)i * 2 + 0] = v0;
  out[(size_t)i * 2 + 1] = w0;
}

extern "C" void kernel_launch(void* const* d_in, const int* in_sizes, int n_in,
                              void* d_out, int out_size, void* d_ws, size_t ws_size,
                              hipStream_t stream) {
  const float* init_state = (const float*)d_in[0];
  const float* scan       = (const float*)d_in[1];
  const float* target     = (const float*)d_in[2];
  const float* wd         = (const float*)d_in[3];
  const float* wo         = (const float*)d_in[4];
  const float* wa         = (const float*)d_in[5];
  const float* wv         = (const float*)d_in[6];
  const float* ww         = (const float*)d_in[7];
  const int B = in_sizes[0] / 3;
  const float upd = 0.2f / (float)B;          // LR_INNER / B (grad of cost.mean())
  const int grid = (B + BLK - 1) / BLK;
  mpc_kernel<<<grid, BLK, 0, stream>>>(init_state, scan, target, wd, wo, wa, wv, ww,
                                       (float*)d_out, B, upd);
}